// KolmogorovArnoldLayer_84928683311874
// MI455X (gfx1250) — compile-verified
//
#include <hip/hip_runtime.h>

typedef float v2f __attribute__((ext_vector_type(2)));
typedef float v8f __attribute__((ext_vector_type(8)));

#define B_ROWS 8192
#define D_FEAT 256
#define N_INNER 64
#define OUT_F 256

// Packed 2xFMA on CDNA5 VOP3P: d = a*b + c  (guaranteed v_pk_fma_f32)
__device__ __forceinline__ v2f pk_fma(v2f a, v2f b, v2f c) {
  v2f d;
  asm("v_pk_fma_f32 %0, %1, %2, %3" : "=v"(d) : "v"(a), "v"(b), "v"(c));
  return d;
}

// ---------------------------------------------------------------------------
// Kernel 1: per-feature tiny MLPs  inner[b,d] = sum_i relu(x*w1+b1)*w2 + b2
// 256 threads = 64 features x 4 i-chunks of 16. Weights register-cached as
// v2f pairs; the i-loop runs on V_PK_FMA_F32 (2 FMA/instr) with two
// independent accumulator chains. ReLU = two dual-issueable v_max_f32.
// 4-lane fold via wave32 shfl_xor.
// ---------------------------------------------------------------------------
__global__ __launch_bounds__(256) void kan_inner_kernel(
    const float* __restrict__ x,  const float* __restrict__ w1,
    const float* __restrict__ b1, const float* __restrict__ w2,
    const float* __restrict__ b2, float* __restrict__ inner)
{
  const int t  = threadIdx.x;
  const int dl = t >> 2;                       // feature within block: 0..63
  const int kq = t & 3;                        // i-chunk: 0..3
  const int d  = blockIdx.x * 64 + dl;         // global feature
  const int ib = kq * 16;                      // first inner index

  // Register-cache 16 (w1,b1,w2) triples as 8 v2f pairs each (b128 loads).
  v2f w1p[8], b1p[8], w2p[8];
  const float4* w1v = (const float4*)(w1 + d * N_INNER + ib);
  const float4* b1v = (const float4*)(b1 + d * N_INNER + ib);
  const float4* w2v = (const float4*)(w2 + d * N_INNER + ib);
#pragma unroll
  for (int j = 0; j < 4; ++j) {
    float4 a = w1v[j], b = b1v[j], c = w2v[j];
    w1p[2*j+0] = (v2f){a.x, a.y}; w1p[2*j+1] = (v2f){a.z, a.w};
    b1p[2*j+0] = (v2f){b.x, b.y}; b1p[2*j+1] = (v2f){b.z, b.w};
    w2p[2*j+0] = (v2f){c.x, c.y}; w2p[2*j+1] = (v2f){c.z, c.w};
  }
  const float b2v = b2[d];
  const v2f zero = (v2f){0.0f, 0.0f};

  const int r0 = blockIdx.y * 128;
#pragma unroll 4
  for (int r = r0; r < r0 + 128; ++r) {
    const float xs = x[r * D_FEAT + d];
    const v2f xv = (v2f){xs, xs};
    v2f acc0 = zero, acc1 = zero;              // 2 independent chains
#pragma unroll
    for (int j = 0; j < 4; ++j) {
      v2f h0 = pk_fma(xv, w1p[2*j+0], b1p[2*j+0]);
      v2f h1 = pk_fma(xv, w1p[2*j+1], b1p[2*j+1]);
      h0 = __builtin_elementwise_max(h0, zero);
      h1 = __builtin_elementwise_max(h1, zero);
      acc0 = pk_fma(h0, w2p[2*j+0], acc0);
      acc1 = pk_fma(h1, w2p[2*j+1], acc1);
    }
    float a = (acc0.x + acc0.y) + (acc1.x + acc1.y);
    // fold the 4 i-chunk lanes (lanes kq=0..3 share a feature)
    a += __shfl_xor(a, 1, 32);
    a += __shfl_xor(a, 2, 32);
    if (kq == 0) inner[r * D_FEAT + d] = a + b2v;
  }
}

// ---------------------------------------------------------------------------
// Kernel 2: outer MLP via V_WMMA_F32_16X16X4_F32.
// One wave owns a 16-row tile; 2 waves/block.
//   GEMM1: [16,256] @ wo1^T[256,64]  -> relu(+bo1) -> g (LDS, stride 68)
//   GEMM2: [16,64]  @ wo2^T[64,256]  -> +bo2 -> out
// A 16x4 f32 frag: lane<16 holds {A[l,K],A[l,K+1]}, lane>=16 {A[l-16,K+2],K+3}
// B 4x16 f32 frag: lane<16 holds {B[K,n],B[K+1,n]}, lane>=16 {B[K+2,n],B[K+3,n]}
// Both are contiguous float2 per lane in row-major storage.
// ---------------------------------------------------------------------------
__global__ __launch_bounds__(64) void kan_outer_kernel(
    const float* __restrict__ inner, const float* __restrict__ wo1,
    const float* __restrict__ bo1,   const float* __restrict__ wo2,
    const float* __restrict__ bo2,   float* __restrict__ out)
{
  __shared__ float lds_g[2][16 * 68];          // stride 68: conflict-free
  const int lane = threadIdx.x & 31;
  const int wid  = threadIdx.x >> 5;
  const int row0 = (blockIdx.x * 2 + wid) * 16;
  const int lr   = lane & 15;                  // in-tile row/col
  const int lh   = lane >> 4;                  // K-half select (0/1)

  // ---------------- GEMM1: 4 independent N-tile accumulator chains ---------
  v8f c0 = {}, c1 = {}, c2 = {}, c3 = {};
  const float* aBase = inner + (size_t)(row0 + lr) * D_FEAT + 2 * lh;
  const float* bB0 = wo1 + (size_t)( 0 + lr) * D_FEAT + 2 * lh;
  const float* bB1 = wo1 + (size_t)(16 + lr) * D_FEAT + 2 * lh;
  const float* bB2 = wo1 + (size_t)(32 + lr) * D_FEAT + 2 * lh;
  const float* bB3 = wo1 + (size_t)(48 + lr) * D_FEAT + 2 * lh;
  for (int k = 0; k < D_FEAT; k += 4) {
    v2f a  = *(const v2f*)(aBase + k);
    v2f b0 = *(const v2f*)(bB0 + k);
    v2f b1 = *(const v2f*)(bB1 + k);
    v2f b2 = *(const v2f*)(bB2 + k);
    v2f b3 = *(const v2f*)(bB3 + k);
    c0 = __builtin_amdgcn_wmma_f32_16x16x4_f32(false, a, false, b0, (short)0, c0, false, false);
    c1 = __builtin_amdgcn_wmma_f32_16x16x4_f32(false, a, false, b1, (short)0, c1, false, false);
    c2 = __builtin_amdgcn_wmma_f32_16x16x4_f32(false, a, false, b2, (short)0, c2, false, false);
    c3 = __builtin_amdgcn_wmma_f32_16x16x4_f32(false, a, false, b3, (short)0, c3, false, false);
  }

  // bias + relu, stage g[16][64] into LDS (C-layout scatter, conflict-free)
  float* g = &lds_g[wid][0];
  const float bv0 = bo1[ 0 + lr], bv1 = bo1[16 + lr];
  const float bv2 = bo1[32 + lr], bv3 = bo1[48 + lr];
#pragma unroll
  for (int j = 0; j < 8; ++j) {
    const int m = j + 8 * lh;
    g[m * 68 +  0 + lr] = fmaxf(c0[j] + bv0, 0.0f);
    g[m * 68 + 16 + lr] = fmaxf(c1[j] + bv1, 0.0f);
    g[m * 68 + 32 + lr] = fmaxf(c2[j] + bv2, 0.0f);
    g[m * 68 + 48 + lr] = fmaxf(c3[j] + bv3, 0.0f);
  }
  __syncthreads();

  // ---------------- GEMM2: cache all 16 A-fragments in registers -----------
  v2f a2[16];
#pragma unroll
  for (int kk = 0; kk < 16; ++kk)
    a2[kk] = *(const v2f*)(g + lr * 68 + 4 * kk + 2 * lh);

  for (int o0 = 0; o0 < OUT_F; o0 += 16) {
    v8f acc = {};
    const float* bB = wo2 + (size_t)(o0 + lr) * N_INNER + 2 * lh;
#pragma unroll
    for (int kk = 0; kk < 16; ++kk) {
      v2f bb = *(const v2f*)(bB + 4 * kk);
      acc = __builtin_amdgcn_wmma_f32_16x16x4_f32(false, a2[kk], false, bb, (short)0, acc, false, false);
    }
    const float bv = bo2[o0 + lr];
#pragma unroll
    for (int j = 0; j < 8; ++j)
      out[(size_t)(row0 + j + 8 * lh) * OUT_F + o0 + lr] = acc[j] + bv;
  }
}

// ---------------------------------------------------------------------------
extern "C" void kernel_launch(void* const* d_in, const int* in_sizes, int n_in,
                              void* d_out, int out_size, void* d_ws, size_t ws_size,
                              hipStream_t stream) {
  const float* x   = (const float*)d_in[0];
  const float* w1  = (const float*)d_in[1];
  const float* b1  = (const float*)d_in[2];
  const float* w2  = (const float*)d_in[3];
  const float* b2  = (const float*)d_in[4];
  const float* wo1 = (const float*)d_in[5];
  const float* bo1 = (const float*)d_in[6];
  const float* wo2 = (const float*)d_in[7];
  const float* bo2 = (const float*)d_in[8];
  float* out   = (float*)d_out;
  float* inner = (float*)d_ws;                 // B*D*4 = 8 MB scratch

  dim3 g1(D_FEAT / 64, B_ROWS / 128);          // (4, 64) = 256 blocks
  kan_inner_kernel<<<g1, 256, 0, stream>>>(x, w1, b1, w2, b2, inner);

  // 256 blocks x 2 waves x 16 rows = 8192 rows
  kan_outer_kernel<<<B_ROWS / 32, 64, 0, stream>>>(inner, wo1, bo1, wo2, bo2, out);
}